// GATv2Model_44796508897977
// MI455X (gfx1250) — compile-verified
//
#include <hip/hip_runtime.h>
#include <hip/hip_bf16.h>
#include <stdint.h>

#define N_NODES 80000
#define N_EDGES 1280000
#define FIN_D   128
#define H_D     64
#define OUT_D   10
#define E_TOT   (N_EDGES + N_NODES)
#define NEG_SLOPE 0.2f

typedef float v2f __attribute__((ext_vector_type(2)));
typedef float v8f __attribute__((ext_vector_type(8)));

// ---- order-preserving uint encoding of float (for atomic segment-max) ----
__device__ __forceinline__ uint32_t f2ord(float f) {
  uint32_t u = __float_as_uint(f);
  return (u & 0x80000000u) ? ~u : (u | 0x80000000u);
}
__device__ __forceinline__ float ord2f(uint32_t k) {
  return (k & 0x80000000u) ? __uint_as_float(k & 0x7FFFFFFFu)
                           : __uint_as_float(~k);
}

// ============================================================================
// Dual projection: xl = h @ Wl^T + bl ; xr = h @ Wr^T + br
// One wave per 16x16 output tile; A-tile (h rows) loaded ONCE and fed to two
// V_WMMA_F32_16X16X4_F32 accumulators (one per weight matrix).
// A per lane: row = lane&15, K = {2*kh, 2*kh+1}, kh = lane>>4
// C/D per lane: col = lane&15, VGPR r -> row r + 8*kh
// ============================================================================
__global__ void gemm_dual_wmma_f32(const float* __restrict__ hsrc, int fin,
                                   const float* __restrict__ Wl,
                                   const float* __restrict__ bl,
                                   const float* __restrict__ Wr,
                                   const float* __restrict__ br,
                                   float* __restrict__ xl,
                                   float* __restrict__ xr, int rowTiles) {
  const int wave = blockIdx.x * (blockDim.x >> 5) + (threadIdx.x >> 5);
  const int lane = threadIdx.x & 31;
  const int numTiles = rowTiles * (H_D / 16);
  if (wave >= numTiles) return;                 // wave-uniform: EXEC stays full
  const int rt = wave / (H_D / 16);
  const int ct = wave % (H_D / 16);
  const int mn = lane & 15;                     // A-row index == B/C col index
  const int kh = lane >> 4;

  const float* arow  = hsrc + (size_t)(rt * 16 + mn) * fin;
  const float* wlrow = Wl   + (size_t)(ct * 16 + mn) * fin;  // B[k][n] = W[n][k]
  const float* wrrow = Wr   + (size_t)(ct * 16 + mn) * fin;

  v8f cl, cr;
  const float bcl = bl[ct * 16 + mn];           // bias depends on column only
  const float bcr = br[ct * 16 + mn];
  #pragma unroll
  for (int r = 0; r < 8; ++r) { cl[r] = bcl; cr[r] = bcr; }

  for (int k0 = 0; k0 < fin; k0 += 4) {
    v2f a  = *(const v2f*)(arow  + k0 + kh * 2);
    v2f bL = *(const v2f*)(wlrow + k0 + kh * 2);
    v2f bR = *(const v2f*)(wrrow + k0 + kh * 2);
    cl = __builtin_amdgcn_wmma_f32_16x16x4_f32(false, a, false, bL,
                                               (short)0, cl, false, false);
    cr = __builtin_amdgcn_wmma_f32_16x16x4_f32(false, a, false, bR,
                                               (short)0, cr, false, false);
  }

  const size_t outBase = (size_t)(rt * 16) * H_D + ct * 16 + mn;
  #pragma unroll
  for (int r = 0; r < 8; ++r) {
    xl[outBase + (size_t)(r + 8 * kh) * H_D] = cl[r];
    xr[outBase + (size_t)(r + 8 * kh) * H_D] = cr[r];
  }
}

// ============================================================================
// Per-layer init (float4-wide): zero accum, zero denom, segment-max = -inf
// ============================================================================
__global__ void layer_init(float4* __restrict__ accum4,
                           float* __restrict__ denom,
                           uint32_t* __restrict__ mord) {
  const int i = blockIdx.x * blockDim.x + threadIdx.x;
  if (i < N_NODES * (H_D / 4)) accum4[i] = make_float4(0.f, 0.f, 0.f, 0.f);
  if (i < N_NODES) { denom[i] = 0.0f; mord[i] = 0x007FFFFFu; /* f2ord(-inf) */ }
}

// ============================================================================
// Pass A: e_j = leaky_relu(xl[s] + xr[d]) . att ; atomic segment-max over dst
// 16 lanes per edge, float4 per lane -> global_load_b128 gathers (L2-resident)
// ============================================================================
__global__ void edge_score(const float* __restrict__ xl,
                           const float* __restrict__ xr,
                           const int* __restrict__ src,
                           const int* __restrict__ dst,
                           const float* __restrict__ att,
                           float* __restrict__ e, uint32_t* __restrict__ mord) {
  const int tid  = blockIdx.x * blockDim.x + threadIdx.x;
  const int edge = tid >> 4;                    // 16 lanes per edge
  const int li   = tid & 15;
  if (edge >= E_TOT) return;
  int s, d;
  if (edge < N_EDGES) { s = src[edge]; d = dst[edge]; }
  else                { s = d = edge - N_EDGES; }          // implicit self-loop

  const float4 a  = *(const float4*)(xl + (size_t)s * H_D + li * 4);
  const float4 b  = *(const float4*)(xr + (size_t)d * H_D + li * 4);
  const float4 av = *(const float4*)(att + li * 4);
  float t0 = a.x + b.x, t1 = a.y + b.y, t2 = a.z + b.z, t3 = a.w + b.w;
  t0 = (t0 > 0.f) ? t0 : NEG_SLOPE * t0;
  t1 = (t1 > 0.f) ? t1 : NEG_SLOPE * t1;
  t2 = (t2 > 0.f) ? t2 : NEG_SLOPE * t2;
  t3 = (t3 > 0.f) ? t3 : NEG_SLOPE * t3;
  float p = t0 * av.x + t1 * av.y + t2 * av.z + t3 * av.w;
  #pragma unroll
  for (int off = 8; off > 0; off >>= 1)         // xor stays within 16-lane group
    p += __shfl_xor(p, off, 32);
  if (li == 0) {
    e[edge] = p;
    atomicMax(mord + d, f2ord(p));
  }
}

// ============================================================================
// Pass B: ex_j = exp(e_j - m[d]); denom[d] += ex_j   (thread per edge)
// ============================================================================
__global__ void edge_exp(float* __restrict__ e,
                         const uint32_t* __restrict__ mord,
                         const int* __restrict__ dst,
                         float* __restrict__ denom) {
  const int j = blockIdx.x * blockDim.x + threadIdx.x;
  if (j >= E_TOT) return;
  const int d = (j < N_EDGES) ? dst[j] : (j - N_EDGES);
  const float ex = __expf(e[j] - ord2f(mord[d]));
  e[j] = ex;                                    // overwrite score with exp
  atomicAdd(denom + d, ex);
}

// ============================================================================
// Pass C: accum[d] += (ex_j / denom[d]) * xl[s]
// 16 lanes per edge; b128 gather of xl row, 4 f32 L2 atomics per lane.
// ============================================================================
__global__ void edge_scatter(const float* __restrict__ ex,
                             const float* __restrict__ denom,
                             const float* __restrict__ xl,
                             const int* __restrict__ src,
                             const int* __restrict__ dst,
                             float* __restrict__ accum) {
  const int tid  = blockIdx.x * blockDim.x + threadIdx.x;
  const int edge = tid >> 4;
  const int li   = tid & 15;
  if (edge >= E_TOT) return;
  int s, d;
  if (edge < N_EDGES) { s = src[edge]; d = dst[edge]; }
  else                { s = d = edge - N_EDGES; }

  const float alpha = ex[edge] / denom[d];
  const float4 v = *(const float4*)(xl + (size_t)s * H_D + li * 4);
  float* out = accum + (size_t)d * H_D + li * 4;
  atomicAdd(out + 0, alpha * v.x);
  atomicAdd(out + 1, alpha * v.y);
  atomicAdd(out + 2, alpha * v.z);
  atomicAdd(out + 3, alpha * v.w);
}

// ============================================================================
// h_out = relu(accum + b)  (in place, float4-wide)
// ============================================================================
__global__ void bias_relu(float4* __restrict__ accum4,
                          const float4* __restrict__ b4) {
  const int i = blockIdx.x * blockDim.x + threadIdx.x;
  if (i >= N_NODES * (H_D / 4)) return;
  const float4 bv = b4[i & (H_D / 4 - 1)];
  float4 v = accum4[i];
  v.x += bv.x; v.y += bv.y; v.z += bv.z; v.w += bv.w;
  v.x = (v.x > 0.f) ? v.x : 0.f;
  v.y = (v.y > 0.f) ? v.y : 0.f;
  v.z = (v.z > 0.f) ? v.z : 0.f;
  v.w = (v.w > 0.f) ? v.w : 0.f;
  accum4[i] = v;
}

// ============================================================================
// out[N,10] = h[N,64] @ Wro[10,64]^T + bro
// ============================================================================
__global__ void readout(const float* __restrict__ h,
                        const float* __restrict__ Wro,
                        const float* __restrict__ bro,
                        float* __restrict__ out) {
  const int t = blockIdx.x * blockDim.x + threadIdx.x;
  const int n = t >> 4;
  const int o = t & 15;
  if (n >= N_NODES || o >= OUT_D) return;
  const float* hr = h + (size_t)n * H_D;
  const float* wr = Wro + o * H_D;
  float acc = bro[o];
  #pragma unroll 8
  for (int k = 0; k < H_D; ++k) acc += hr[k] * wr[k];
  out[(size_t)n * OUT_D + o] = acc;
}

// ============================================================================
extern "C" void kernel_launch(void* const* d_in, const int* in_sizes, int n_in,
                              void* d_out, int out_size, void* d_ws, size_t ws_size,
                              hipStream_t stream) {
  (void)in_sizes; (void)n_in; (void)out_size; (void)ws_size;

  const float* x    = (const float*)d_in[0];
  const int*   eidx = (const int*)d_in[1];
  const int*   src  = eidx;            // edge_index[0]
  const int*   dst  = eidx + N_EDGES;  // edge_index[1]
  // d_in[2] = batch (unused for node-level readout)
  const float* Wro = (const float*)d_in[21];
  const float* bro = (const float*)d_in[22];
  float* outp = (float*)d_out;

  // ---- workspace partition (floats) ----
  const size_t NH = (size_t)N_NODES * H_D;
  float* ws   = (float*)d_ws;
  float* xl   = ws;                 // N*64
  float* xr   = xl + NH;            // N*64
  float* h0   = xr + NH;            // N*64  activation ping
  float* h1   = h0 + NH;            // N*64  activation pong
  float* e    = h1 + NH;            // E_TOT
  float* den  = e + E_TOT;          // N
  uint32_t* mo = (uint32_t*)(den + N_NODES);  // N

  const int rowTiles  = N_NODES / 16;              // 5000
  const int gemmWaves = rowTiles * (H_D / 16);     // 20000 tiles
  const dim3 blk(256);
  const dim3 gemmGrid((gemmWaves + 7) / 8);        // 8 waves per 256-thr block
  const dim3 init4Grid((N_NODES * (H_D / 4) + 255) / 256);
  const dim3 edge16Grid(((size_t)E_TOT * 16 + 255) / 256);  // 16 lanes/edge
  const dim3 edgeThrGrid((E_TOT + 255) / 256);
  const dim3 roGrid((N_NODES * 16 + 255) / 256);

  const float* hin = x;
  int fin = FIN_D;
  for (int l = 0; l < 3; ++l) {
    const int base = 3 + 6 * l;
    const float* Wl  = (const float*)d_in[base + 0];
    const float* bl  = (const float*)d_in[base + 1];
    const float* Wr  = (const float*)d_in[base + 2];
    const float* br  = (const float*)d_in[base + 3];
    const float* att = (const float*)d_in[base + 4];
    const float* bb  = (const float*)d_in[base + 5];
    float* hout = (l & 1) ? h1 : h0;

    gemm_dual_wmma_f32<<<gemmGrid, blk, 0, stream>>>(hin, fin, Wl, bl, Wr, br,
                                                     xl, xr, rowTiles);
    layer_init<<<init4Grid, blk, 0, stream>>>((float4*)hout, den, mo);
    edge_score<<<edge16Grid, blk, 0, stream>>>(xl, xr, src, dst, att, e, mo);
    edge_exp<<<edgeThrGrid, blk, 0, stream>>>(e, mo, dst, den);
    edge_scatter<<<edge16Grid, blk, 0, stream>>>(e, den, xl, src, dst, hout);
    bias_relu<<<init4Grid, blk, 0, stream>>>((float4*)hout, (const float4*)bb);

    hin = hout;
    fin = H_D;
  }

  readout<<<roGrid, blk, 0, stream>>>(hin, Wro, bro, outp);
}